// color_temp_tuning_net_45011257262170
// MI455X (gfx1250) — compile-verified
//
#include <hip/hip_runtime.h>

typedef __attribute__((ext_vector_type(2))) float v2f;
typedef __attribute__((ext_vector_type(8))) float v8f;

static __device__ __forceinline__ v2f mk2(float x, float y) {
  v2f v; v.x = x; v.y = y; return v;
}

// ---------------------------------------------------------------------------
// Kernel 1: tiny MLP producing per-batch 3x34 weight matrix into workspace.
// One block per batch. One-hot input => layer 1 is a 2-column gather of w1.
// ---------------------------------------------------------------------------
__global__ __launch_bounds__(256) void mlp_kernel(
    const int* __restrict__ src, const int* __restrict__ tgt,
    const float* __restrict__ w1, const float* __restrict__ a1,
    const float* __restrict__ w2, const float* __restrict__ a2,
    const float* __restrict__ w3, float* __restrict__ wout)
{
  const int b = blockIdx.x;
  const int t = threadIdx.x;
  __shared__ float h1[128];
  __shared__ float h2[256];
  const float al1 = a1[0];
  const float al2 = a2[0];
  const int s  = src[b];
  const int tg = tgt[b];

  if (t < 128) {
    float v = w1[t * 10 + s] + w1[t * 10 + 5 + tg];   // label @ w1.T (one-hot)
    h1[t] = fmaxf(v, 0.f) + al1 * fminf(v, 0.f);      // prelu
  }
  __syncthreads();

  {
    const float* w2r = w2 + t * 128;
    float acc = 0.f;
#pragma unroll 4
    for (int j = 0; j < 128; ++j) acc = fmaf(w2r[j], h1[j], acc);
    h2[t] = fmaxf(acc, 0.f) + al2 * fminf(acc, 0.f);  // prelu
  }
  __syncthreads();

  if (t < 102) {
    const float* w3r = w3 + t * 256;
    float acc = 0.f;
#pragma unroll 4
    for (int k = 0; k < 256; ++k) acc = fmaf(w3r[k], h2[k], acc);
    wout[b * 102 + t] = acc;                          // weight[b][3][34] flat
  }
}

// ---------------------------------------------------------------------------
// Kernel 2: per-pixel poly34 terms + 34->3 projection via V_WMMA_F32_16X16X4_F32.
// Wave handles 16 pixels/iter: lanes n and n+16 both own pixel n (hi half
// supplies K=2,3 of each 4-term chunk per the f32 A/B VGPR layout).
// A = per-batch weights (M=3 rows used), B = terms, D row k = channel k.
// ---------------------------------------------------------------------------
#define WAVES_PER_BLOCK 8
#define ITERS 32

__global__ __launch_bounds__(256) void apply_kernel(
    const float* __restrict__ rgb, const float* __restrict__ wts,
    float* __restrict__ out, int HW)
{
  const int b    = blockIdx.y;
  const int lane = (int)(threadIdx.x & 31u);
  const int wave = (int)(threadIdx.x >> 5u);
  const int m    = lane & 15;
  const bool hi  = lane >= 16;

  // Preload A operands: 9 chunks of K=4 (34 terms padded to 36). Branch-free:
  // clamped-index loads + cndmask zeroing (rows m>=3 and cols >=34 are 0).
  const float* __restrict__ W = wts + b * 102;       // [3][34]
  const int row = (m < 3) ? m : 0;
  v2f A[9];
#pragma unroll
  for (int j = 0; j < 9; ++j) {
    const int c0 = 4 * j + (hi ? 2 : 0);
    const int i0 = (c0     < 34) ? c0     : 0;
    const int i1 = (c0 + 1 < 34) ? c0 + 1 : 0;
    float a0 = W[row * 34 + i0];
    float a1 = W[row * 34 + i1];
    a0 = (m < 3 && c0     < 34) ? a0 : 0.f;
    a1 = (m < 3 && c0 + 1 < 34) ? a1 : 0.f;
    A[j] = mk2(a0, a1);
  }

  const size_t planeOff = (size_t)b * 3 * (size_t)HW;
  const float* Rp = rgb + planeOff;
  const float* Gp = Rp + HW;
  const float* Bp = Gp + HW;
  float* O0 = out + planeOff;
  float* O1 = O0 + HW;
  float* O2 = O1 + HW;

  const unsigned pixBase =
      ((unsigned)blockIdx.x * WAVES_PER_BLOCK + (unsigned)wave) * (ITERS * 16) +
      (unsigned)m;

#pragma unroll 1
  for (unsigned it = 0; it < ITERS; ++it) {
    const unsigned pix = pixBase + it * 16u;   // unsigned -> saddr+voffset form
    const float r  = Rp[pix];
    const float g  = Gp[pix];
    const float bb = Bp[pix];

    // 34 polynomial terms (shared subexpressions, ~28 mults)
    const float r2 = r * r,   g2 = g * g,   b2 = bb * bb;
    const float rg = r * g,   gb = g * bb,  rb = r * bb;
    const float r3 = r2 * r,  g3 = g2 * g,  b3 = b2 * bb;
    const float rg2 = r * g2, gb2 = g * b2, rb2 = r * b2;
    const float gr2 = g * r2, bg2 = bb * g2, br2 = bb * r2;
    const float rgb_ = rg * bb;
    const float r4 = r2 * r2, g4 = g2 * g2, b4 = b2 * b2;
    const float r3g = r3 * g,  r3b = r3 * bb;
    const float g3r = g3 * r,  g3b = g3 * bb;
    const float b3r = b3 * r,  b3g = b3 * g;
    const float r2g2 = r2 * g2, g2b2 = g2 * b2, r2b2 = r2 * b2;
    const float r2gb = r2 * gb, g2rb = g2 * rb, b2rg = b2 * rg;

    v8f acc = {};
    v2f Bv;

    // chunk j: lo lanes carry terms 4j,4j+1 ; hi lanes carry 4j+2,4j+3
    Bv  = hi ? mk2(bb,   r2)   : mk2(r,    g);
    acc = __builtin_amdgcn_wmma_f32_16x16x4_f32(false, A[0], false, Bv, (short)0, acc, false, false);
    Bv  = hi ? mk2(rg,   gb)   : mk2(g2,   b2);
    acc = __builtin_amdgcn_wmma_f32_16x16x4_f32(false, A[1], false, Bv, (short)0, acc, false, false);
    Bv  = hi ? mk2(g3,   b3)   : mk2(rb,   r3);
    acc = __builtin_amdgcn_wmma_f32_16x16x4_f32(false, A[2], false, Bv, (short)0, acc, false, false);
    Bv  = hi ? mk2(rb2,  gr2)  : mk2(rg2,  gb2);
    acc = __builtin_amdgcn_wmma_f32_16x16x4_f32(false, A[3], false, Bv, (short)0, acc, false, false);
    Bv  = hi ? mk2(rgb_, r4)   : mk2(bg2,  br2);
    acc = __builtin_amdgcn_wmma_f32_16x16x4_f32(false, A[4], false, Bv, (short)0, acc, false, false);
    Bv  = hi ? mk2(r3g,  r3b)  : mk2(g4,   b4);
    acc = __builtin_amdgcn_wmma_f32_16x16x4_f32(false, A[5], false, Bv, (short)0, acc, false, false);
    Bv  = hi ? mk2(b3r,  b3g)  : mk2(g3r,  g3b);
    acc = __builtin_amdgcn_wmma_f32_16x16x4_f32(false, A[6], false, Bv, (short)0, acc, false, false);
    Bv  = hi ? mk2(r2b2, r2gb) : mk2(r2g2, g2b2);
    acc = __builtin_amdgcn_wmma_f32_16x16x4_f32(false, A[7], false, Bv, (short)0, acc, false, false);
    Bv  = hi ? mk2(0.f,  0.f)  : mk2(g2rb, b2rg);
    acc = __builtin_amdgcn_wmma_f32_16x16x4_f32(false, A[8], false, Bv, (short)0, acc, false, false);

    // D: VGPR k, lanes 0-15 = channel k of pixel (lane). Store 3 planes.
    if (!hi) {
      O0[pix] = acc[0];
      O1[pix] = acc[1];
      O2[pix] = acc[2];
    }
    // EXEC reconverges here; next iteration's WMMAs see all-1s EXEC again.
  }
}

// ---------------------------------------------------------------------------
extern "C" void kernel_launch(void* const* d_in, const int* in_sizes, int n_in,
                              void* d_out, int out_size, void* d_ws, size_t ws_size,
                              hipStream_t stream) {
  const int*   src = (const int*)d_in[0];
  const int*   tgt = (const int*)d_in[1];
  const float* rgb = (const float*)d_in[2];
  const float* w1  = (const float*)d_in[3];
  const float* a1  = (const float*)d_in[4];
  const float* w2  = (const float*)d_in[5];
  const float* a2  = (const float*)d_in[6];
  const float* w3  = (const float*)d_in[7];
  float* outp = (float*)d_out;
  float* wts  = (float*)d_ws;   // 8*102 floats of scratch

  const int B  = in_sizes[0];
  const int HW = in_sizes[2] / (3 * B);   // 1024*1024

  mlp_kernel<<<dim3(B), dim3(256), 0, stream>>>(src, tgt, w1, a1, w2, a2, w3, wts);

  const int pixPerBlock = WAVES_PER_BLOCK * ITERS * 16;   // 4096
  dim3 grid(HW / pixPerBlock, B);
  apply_kernel<<<grid, dim3(256), 0, stream>>>(rgb, wts, outp, HW);
}